// HybridrPPGLoss_20246475833391
// MI455X (gfx1250) — compile-verified
//
#include <hip/hip_runtime.h>
#include <hip/hip_bf16.h>
#include <math.h>
#include <stdint.h>

#ifndef __has_builtin
#define __has_builtin(x) 0
#endif

#define NCOL 8192
#define NROW 2048
#define EPS 1e-8f

// gfx1250 async global->LDS staging (ASYNCcnt path). Flip to 0 if asm rejects.
#define USE_ASYNC_LDS 1

typedef __attribute__((ext_vector_type(2))) float v2f;
typedef __attribute__((ext_vector_type(8))) float v8f;

#if __has_builtin(__builtin_amdgcn_wmma_f32_16x16x4_f32)
#define USE_WMMA 1
#else
#define USE_WMMA 0
#endif

__device__ __forceinline__ uint32_t lds_addr32(const void* p) {
  // generic LDS pointer: low 32 bits are the LDS byte offset
  return (uint32_t)(size_t)p;
}

// ---------------------------------------------------------------------------
// Kernel A: per-row stats + 31 circular shift correlations via WMMA f32 16x16x4
// One block per row, 256 threads (8 waves, wave32).
// D[a][b] accumulated over all circular windows equals S(a-b) exactly, so all
// 31 lags live in one 16x16 accumulator tile.
// LDS layout (float lds[16384], 64KB):
//   phase 1/2: sP = lds[0..8191], sT = lds[8192..16383]
//   phase 3 overlay: Dpart lds[0..2047] (WMMA) or accs lds[0..8190] (fallback),
//                    sums lds[8192..9215], Dsum lds[9216..9471], pear lds[9472..]
// ---------------------------------------------------------------------------
__launch_bounds__(256)
__global__ void corr_kernel(const float* __restrict__ preds,
                            const float* __restrict__ targets,
                            float* __restrict__ ws_best) {
  __shared__ __align__(16) float lds[16384];
  const int tid = threadIdx.x;
  const int row = blockIdx.x;
  const float* gp = preds  + (size_t)row * NCOL;
  const float* gt = targets + (size_t)row * NCOL;

  float sp = 0.f, sp2 = 0.f, st = 0.f, st2 = 0.f;
#if USE_ASYNC_LDS
  // ---- stage rows via tensor/async DMA path: global -> LDS, no VGPR bounce
  {
    const float4* gp4 = (const float4*)gp;
    const float4* gt4 = (const float4*)gt;
    uint32_t ldsP = lds_addr32(&lds[0]);
    uint32_t ldsT = lds_addr32(&lds[8192]);
#pragma unroll
    for (int j = 0; j < 8; ++j) {
      int i4 = tid + (j << 8);
      asm volatile("global_load_async_to_lds_b128 %0, %1, off"
                   :: "v"(ldsP + (uint32_t)i4 * 16u), "v"(gp4 + i4) : "memory");
      asm volatile("global_load_async_to_lds_b128 %0, %1, off"
                   :: "v"(ldsT + (uint32_t)i4 * 16u), "v"(gt4 + i4) : "memory");
    }
    asm volatile("s_wait_asynccnt 0x0" ::: "memory");
  }
  __syncthreads();
  // ---- read back for moment sums (LDS-resident now)
  {
    const float4* sP4 = (const float4*)&lds[0];
    const float4* sT4 = (const float4*)&lds[8192];
#pragma unroll
    for (int j = 0; j < 8; ++j) {
      int i4 = tid + (j << 8);
      float4 a = sP4[i4];
      float4 b = sT4[i4];
      sp  += a.x + a.y + a.z + a.w;
      sp2 += a.x * a.x + a.y * a.y + a.z * a.z + a.w * a.w;
      st  += b.x + b.y + b.z + b.w;
      st2 += b.x * b.x + b.y * b.y + b.z * b.z + b.w * b.w;
    }
  }
#else
  {
    const float4* gp4 = (const float4*)gp;
    const float4* gt4 = (const float4*)gt;
    float4* sP4 = (float4*)&lds[0];
    float4* sT4 = (float4*)&lds[8192];
#pragma unroll
    for (int j = 0; j < 8; ++j) {
      int i4 = tid + (j << 8);
      float4 a = gp4[i4];
      float4 b = gt4[i4];
      sP4[i4] = a;
      sT4[i4] = b;
      sp  += a.x + a.y + a.z + a.w;
      sp2 += a.x * a.x + a.y * a.y + a.z * a.z + a.w * a.w;
      st  += b.x + b.y + b.z + b.w;
      st2 += b.x * b.x + b.y * b.y + b.z * b.z + b.w * b.w;
    }
  }
  __syncthreads();
#endif

  const float* sP = &lds[0];
  const float* sT = &lds[8192];

#if USE_WMMA
  const int wave = tid >> 5;
  const int lane = tid & 31;
  const int mn = lane & 15;            // A row (m) == B col (n)
  const int kb = (lane >> 4) << 1;     // K pair base: 0 or 2
  const int base0 = wave << 10;        // each wave covers 1024 elements
  const int fo = base0 + kb + mn;

  v8f acc0 = {0.f, 0.f, 0.f, 0.f, 0.f, 0.f, 0.f, 0.f};
  v8f acc1 = {0.f, 0.f, 0.f, 0.f, 0.f, 0.f, 0.f, 0.f};

  // software-pipelined: fetch chunk pair c+2 while WMMA'ing chunk pair c,
  // so s_wait_dscnt becomes partial instead of a full drain per WMMA.
  v2f a0, b0, a1, b1;
  {
    int i0 = fo;
    a0.x = sT[i0 & 8191]; a0.y = sT[(i0 + 1) & 8191];
    b0.x = sP[i0 & 8191]; b0.y = sP[(i0 + 1) & 8191];
    int i1 = i0 + 4;
    a1.x = sT[i1 & 8191]; a1.y = sT[(i1 + 1) & 8191];
    b1.x = sP[i1 & 8191]; b1.y = sP[(i1 + 1) & 8191];
  }
  for (int c = 0; c < 256; c += 2) {
    int inx = fo + (((c + 2) & 255) << 2);     // wraps to chunk 0 on last iter
    v2f na0, nb0, na1, nb1;
    na0.x = sT[inx & 8191]; na0.y = sT[(inx + 1) & 8191];
    nb0.x = sP[inx & 8191]; nb0.y = sP[(inx + 1) & 8191];
    int iny = inx + 4;
    na1.x = sT[iny & 8191]; na1.y = sT[(iny + 1) & 8191];
    nb1.x = sP[iny & 8191]; nb1.y = sP[(iny + 1) & 8191];
    acc0 = __builtin_amdgcn_wmma_f32_16x16x4_f32(false, a0, false, b0,
                                                 (short)0, acc0, false, false);
    acc1 = __builtin_amdgcn_wmma_f32_16x16x4_f32(false, a1, false, b1,
                                                 (short)0, acc1, false, false);
    a0 = na0; b0 = nb0; a1 = na1; b1 = nb1;
  }
#pragma unroll
  for (int r = 0; r < 8; ++r) acc0[r] += acc1[r];
  __syncthreads();
  {
    const int hh = (lane >> 4) << 3;
#pragma unroll
    for (int r = 0; r < 8; ++r)
      lds[(wave << 8) + ((r + hh) << 4) + mn] = acc0[r];
  }
#else
  float accs[31];
#pragma unroll
  for (int s = 0; s < 31; ++s) accs[s] = 0.f;
  for (int i = tid; i < 8192; i += 256) {
    float pv = sP[i];
#pragma unroll
    for (int s = 0; s < 31; ++s) accs[s] += pv * sT[(i + s - 15) & 8191];
  }
  __syncthreads();
#pragma unroll
  for (int s = 0; s < 31; ++s) lds[(tid << 5) + s] = accs[s];
#endif

  lds[8192 + tid] = sp;
  lds[8448 + tid] = sp2;
  lds[8704 + tid] = st;
  lds[8960 + tid] = st2;
  __syncthreads();

  for (int s2 = 128; s2 > 0; s2 >>= 1) {
    if (tid < s2) {
      lds[8192 + tid] += lds[8192 + tid + s2];
      lds[8448 + tid] += lds[8448 + tid + s2];
      lds[8704 + tid] += lds[8704 + tid + s2];
      lds[8960 + tid] += lds[8960 + tid + s2];
    }
    __syncthreads();
  }

#if USE_WMMA
  {
    float dsum = 0.f;
#pragma unroll
    for (int w = 0; w < 8; ++w) dsum += lds[(w << 8) + tid];
    lds[9216 + tid] = dsum;
  }
  __syncthreads();
#endif

  if (tid < 31) {
    const int s = tid - 15;
#if USE_WMMA
    const int aa = s > 0 ? s : 0;
    const int bb = s > 0 ? 0 : -s;
    float S = lds[9216 + aa * 16 + bb];
#else
    float S = 0.f;
    for (int t = 0; t < 256; ++t) S += lds[(t << 5) + tid];
#endif
    const float Sp = lds[8192], Sp2 = lds[8448];
    const float St = lds[8704], St2 = lds[8960];
    const float invN = 1.0f / 8192.0f;
    float cov  = S - Sp * St * invN;
    float stdp = sqrtf(Sp2 - Sp * Sp * invN + EPS);
    float stdt = sqrtf(St2 - St * St * invN + EPS);
    lds[9472 + tid] = cov / (stdp * stdt);
  }
  __syncthreads();
  if (tid == 0) {
    float best = -1.0f;
    for (int i = 0; i < 31; ++i) best = fmaxf(best, lds[9472 + i]);
    ws_best[row] = best;
  }
}

// ---------------------------------------------------------------------------
// Kernel B: per-row rFFT(8192) power spectra + KL divergence.
// Packed complex FFT-4096, in-place radix-4 DIF in LDS (digit-reversed output
// resolved at the unpack step). Stage twiddles come from an LDS table of
// W_4096 powers (built once per block).
// LDS (float l[14596], ~58KB):
//   re = l[0..4095], im = l[4096..8191]
//   PP (preds power, bins 1..4096) = l[8192 + k]
//   PT (targets power)             = l[k]  (overlays re/im after t-FFT)
//   RED = l[12292..12547], TWC = l[12548..13571], TWS = l[13572..14595]
// ---------------------------------------------------------------------------
__device__ __forceinline__ unsigned dr6(unsigned x) {  // reverse 6 base-4 digits
  unsigned r = 0;
#pragma unroll
  for (int d = 0; d < 6; ++d) { r = (r << 2) | (x & 3u); x >>= 2; }
  return r;
}

__device__ __forceinline__ void fft4096_dif(float* re, float* im,
                                            const float* TWC, const float* TWS,
                                            int tid) {
#pragma unroll
  for (int st = 0; st < 6; ++st) {
    const int Lq = 10 - 2 * st;       // log2(Ns)
    const int Ns = 1 << Lq;
#pragma unroll
    for (int it = 0; it < 4; ++it) {
      int u = tid + (it << 8);        // butterfly index 0..1023
      int j = u & (Ns - 1);
      int b = u >> Lq;
      int base = (b << (Lq + 2)) + j;
      int i0 = base, i1 = base + Ns, i2 = base + 2 * Ns, i3 = base + 3 * Ns;
      float x0r = re[i0], x0i = im[i0];
      float x1r = re[i1], x1i = im[i1];
      float x2r = re[i2], x2i = im[i2];
      float x3r = re[i3], x3i = im[i3];
      float t0r = x0r + x2r, t0i = x0i + x2i;
      float t1r = x0r - x2r, t1i = x0i - x2i;
      float t2r = x1r + x3r, t2i = x1i + x3i;
      float t3r = x1i - x3i, t3i = x3r - x1r;   // -i*(x1-x3)
      float y0r = t0r + t2r, y0i = t0i + t2i;
      float y1r = t1r + t3r, y1i = t1i + t3i;
      float y2r = t0r - t2r, y2i = t0i - t2i;
      float y3r = t1r - t3r, y3i = t1i - t3i;
      int ti = j << (10 - Lq);        // table index: j * 1024/Ns
      float c1 = TWC[ti], s1 = TWS[ti];
      float c2 = c1 * c1 - s1 * s1, s2 = 2.0f * c1 * s1;
      float c3 = c2 * c1 - s2 * s1, s3 = s2 * c1 + c2 * s1;
      re[i0] = y0r;                   im[i0] = y0i;
      re[i1] = y1r * c1 - y1i * s1;   im[i1] = y1r * s1 + y1i * c1;
      re[i2] = y2r * c2 - y2i * s2;   im[i2] = y2r * s2 + y2i * c2;
      re[i3] = y3r * c3 - y3i * s3;   im[i3] = y3r * s3 + y3i * c3;
    }
    __syncthreads();
  }
}

__device__ __forceinline__ float power_pass(const float* re, const float* im,
                                            float* out, int tid) {
  float pw[16];
  float local = 0.f;
  float b4096 = 0.f;
  if (tid == 0) {
    float d = re[0] - im[0];
    b4096 = d * d;
  }
#pragma unroll
  for (int i = 0; i < 8; ++i) {
    int k = 1 + tid + (i << 8);        // 1..2048
    int m = 4096 - k;                  // 2048..4095
    unsigned ik = dr6((unsigned)k), imm = dr6((unsigned)m);
    float ar = re[ik], ai = im[ik];
    float br = re[imm], bi = im[imm];
    float Er = 0.5f * (ar + br), Ei = 0.5f * (ai - bi);
    float Or = 0.5f * (ai + bi), Oi = 0.5f * (br - ar);
    float wc, wsn;
    __sincosf(-7.669903939428206e-4f * (float)k, &wsn, &wc);  // -2*pi*k/8192
    float pr = wc * Or - wsn * Oi;
    float qi = wc * Oi + wsn * Or;
    float xr = Er + pr, xi = Ei + qi;
    pw[2 * i] = xr * xr + xi * xi;                 // bin k
    float yr = Er - pr, yi = Ei - qi;
    pw[2 * i + 1] = (k < 2048) ? (yr * yr + yi * yi) : 0.f;  // bin 4096-k
    local += pw[2 * i] + pw[2 * i + 1];
  }
  __syncthreads();
#pragma unroll
  for (int i = 0; i < 8; ++i) {
    int k = 1 + tid + (i << 8);
    out[k] = pw[2 * i];
    if (k < 2048) out[4096 - k] = pw[2 * i + 1];
  }
  if (tid == 0) {
    out[4096] = b4096;
    local += b4096;
  }
  return local;
}

__device__ __forceinline__ void pack_row(const float* __restrict__ g,
                                         float* re, float* im, int tid) {
#if USE_ASYNC_LDS
  // async b32 loads deinterleave even/odd samples straight into re[]/im[]
  uint32_t aRe = lds_addr32(re);
  uint32_t aIm = lds_addr32(im);
#pragma unroll
  for (int i = 0; i < 16; ++i) {
    int n = tid + (i << 8);
    asm volatile("global_load_async_to_lds_b32 %0, %1, off"
                 :: "v"(aRe + (uint32_t)n * 4u), "v"(g + 2 * n) : "memory");
    asm volatile("global_load_async_to_lds_b32 %0, %1, off"
                 :: "v"(aIm + (uint32_t)n * 4u), "v"(g + 2 * n + 1) : "memory");
  }
  asm volatile("s_wait_asynccnt 0x0" ::: "memory");
#else
  const float2* g2 = (const float2*)g;
#pragma unroll
  for (int i = 0; i < 16; ++i) {
    int n = tid + (i << 8);
    float2 z = g2[n];
    re[n] = z.x;
    im[n] = z.y;
  }
#endif
}

__launch_bounds__(256)
__global__ void fft_kl_kernel(const float* __restrict__ preds,
                              const float* __restrict__ targets,
                              float* __restrict__ ws_kl) {
  __shared__ __align__(16) float l[14596];
  const int tid = threadIdx.x;
  const int row = blockIdx.x;
  const float* gp = preds  + (size_t)row * NCOL;
  const float* gt = targets + (size_t)row * NCOL;
  float* re  = &l[0];
  float* im  = &l[4096];
  float* PP  = &l[8192];
  float* PT  = &l[0];
  float* RED = &l[12292];
  float* TWC = &l[12548];
  float* TWS = &l[13572];

  // twiddle table: W_4096^i, i < 1024
#pragma unroll
  for (int i2 = 0; i2 < 4; ++i2) {
    int i = tid + (i2 << 8);
    float cc, ss;
    __sincosf(-1.5339807878856412e-3f * (float)i, &ss, &cc);  // -2*pi/4096
    TWC[i] = cc;
    TWS[i] = ss;
  }

  // ---- preds FFT + power
  pack_row(gp, re, im, tid);
  __syncthreads();
  fft4096_dif(re, im, TWC, TWS, tid);
  float lp = power_pass(re, im, PP, tid);
  RED[tid] = lp;
  __syncthreads();
  for (int s2 = 128; s2 > 0; s2 >>= 1) {
    if (tid < s2) RED[tid] += RED[tid + s2];
    __syncthreads();
  }
  const float Sp = RED[0];
  __syncthreads();

  // ---- targets FFT + power
  pack_row(gt, re, im, tid);
  __syncthreads();
  fft4096_dif(re, im, TWC, TWS, tid);
  float lt = power_pass(re, im, PT, tid);
  RED[tid] = lt;
  __syncthreads();
  for (int s2 = 128; s2 > 0; s2 >>= 1) {
    if (tid < s2) RED[tid] += RED[tid + s2];
    __syncthreads();
  }
  const float St = RED[0];
  __syncthreads();

  // ---- KL(prob_t || prob_p)
  const float invSp = 1.0f / (Sp + EPS);
  const float invSt = 1.0f / (St + EPS);
  float kl = 0.f;
#pragma unroll
  for (int i = 0; i < 16; ++i) {
    int k = 1 + tid + (i << 8);
    float pt = PT[k] * invSt;
    float pp = PP[k] * invSp;
    kl += pt * (__logf(pt + EPS) - __logf(pp + EPS));
  }
  RED[tid] = kl;
  __syncthreads();
  for (int s2 = 128; s2 > 0; s2 >>= 1) {
    if (tid < s2) RED[tid] += RED[tid + s2];
    __syncthreads();
  }
  if (tid == 0) ws_kl[row] = RED[0];
}

// ---------------------------------------------------------------------------
// Kernel C: final reduction -> scalar loss
// ---------------------------------------------------------------------------
__launch_bounds__(256)
__global__ void finalize_kernel(const float* __restrict__ wsv,
                                float* __restrict__ out) {
  __shared__ float red[512];
  const int tid = threadIdx.x;
  float sb = 0.f, sk = 0.f;
  for (int i = tid; i < NROW; i += 256) {
    sb += wsv[i];
    sk += wsv[NROW + i];
  }
  red[tid] = sb;
  red[256 + tid] = sk;
  __syncthreads();
  for (int s2 = 128; s2 > 0; s2 >>= 1) {
    if (tid < s2) {
      red[tid] += red[tid + s2];
      red[256 + tid] += red[256 + tid + s2];
    }
    __syncthreads();
  }
  if (tid == 0) {
    float time_loss = 1.0f - red[0] * (1.0f / (float)NROW);
    float freq_loss = red[256] * (1.0f / (float)NROW);
    out[0] = 0.5f * time_loss + 0.5f * freq_loss;
  }
}

extern "C" void kernel_launch(void* const* d_in, const int* in_sizes, int n_in,
                              void* d_out, int out_size, void* d_ws, size_t ws_size,
                              hipStream_t stream) {
  const float* preds   = (const float*)d_in[0];
  const float* targets = (const float*)d_in[1];
  float* ws = (float*)d_ws;            // [0..2047] best_pearson, [2048..4095] kl
  corr_kernel<<<NROW, 256, 0, stream>>>(preds, targets, ws);
  fft_kl_kernel<<<NROW, 256, 0, stream>>>(preds, targets, ws + NROW);
  finalize_kernel<<<1, 256, 0, stream>>>(ws, (float*)d_out);
}